// GraphSAGE_NET_19542101197287
// MI455X (gfx1250) — compile-verified
//
#include <hip/hip_runtime.h>

// GraphSAGE (2-layer, mean aggregation) for MI455X / gfx1250, wave32.
//
// Pipeline (all on `stream`):
//   zero(agg, cnt)
//   pack W1_l/W1_r/W2_l/W2_r into WMMA-B fragment order (W2 zero-padded to 48 cols)
//   edge_scatter: agg += x[src] (f32 atomics), cnt += 1
//   sage_wmma<8,128,relu>:  h  = relu((agg/cnt) @ W1_l + x @ W1_r + b1)   [v_wmma_f32_16x16x4_f32]
//   zero(agg)
//   edge_scatter: agg += h[src]
//   sage_wmma<3,40,none>:   out = (agg/cnt) @ W2_l + h @ W2_r + b2
//
// WMMA f32 16x16x4 operand layout (ISA 7.12.2, wave32):
//   A (16x4, MxK): lane L holds row M=L&15; VGPR0/1 hold K=koff,koff+1 where koff = (L>=16)?2:0
//   B (4x16, KxN): lane L holds col N=L&15; VGPR0/1 hold K=koff,koff+1 (same half rule)
//   C/D (16x16):   VGPR r, lanes 0-15 -> M=r, lanes 16-31 -> M=r+8; N = lane&15

#define N_NODES 100000
#define N_EDGES 800000
#define IN_DIM  128
#define HIDDEN  128
#define CLASSES 40

typedef __attribute__((ext_vector_type(2))) float v2f;
typedef __attribute__((ext_vector_type(8))) float v8f;

__global__ __launch_bounds__(256) void zero_f32(float* __restrict__ p, long long n) {
  long long i = (long long)blockIdx.x * blockDim.x + threadIdx.x;
  long long stride = (long long)gridDim.x * blockDim.x;
  for (; i < n; i += stride) p[i] = 0.0f;
}

// One wave32 per edge: lane loads float4 of source row (global_load_b128),
// 4x global_atomic_add_f32 into destination row. Counts added only on pass 1.
__global__ __launch_bounds__(256) void edge_scatter(const float* __restrict__ feat,
                                                    const int* __restrict__ ei,
                                                    float* __restrict__ agg,
                                                    float* __restrict__ cnt,
                                                    int add_cnt) {
  int wave = (int)((blockIdx.x * blockDim.x + threadIdx.x) >> 5);
  int lane = threadIdx.x & 31;
  if (wave >= N_EDGES) return;                 // wave-uniform
  int src = ei[wave];
  int dst = ei[N_EDGES + wave];
  const float4 v = ((const float4*)(feat + (size_t)src * IN_DIM))[lane];
  float* d = agg + (size_t)dst * IN_DIM + lane * 4;
  atomicAdd(d + 0, v.x);
  atomicAdd(d + 1, v.y);
  atomicAdd(d + 2, v.z);
  atomicAdd(d + 3, v.w);
  if (add_cnt && lane == 0) atomicAdd(cnt + dst, 1.0f);
}

// Repack W[K][Nsrc] (row-major) into per-(kstep,ntile) B fragments so the GEMM's
// inner-loop B load is a single coalesced b64 per lane. Columns >= Nsrc -> 0.
__global__ __launch_bounds__(256) void pack_w(const float* __restrict__ W, int K, int Nsrc,
                                              int NT, v2f* __restrict__ out) {
  int idx = blockIdx.x * blockDim.x + threadIdx.x;
  int total = (K >> 2) * NT * 32;
  if (idx >= total) return;
  int lane = idx & 31;
  int nt = (idx >> 5) % NT;
  int ks = (idx >> 5) / NT;
  int kk = ks * 4 + ((lane >> 4) << 1);
  int n = nt * 16 + (lane & 15);
  v2f r;
  if (n < Nsrc) { r.x = W[kk * Nsrc + n]; r.y = W[(kk + 1) * Nsrc + n]; }
  else          { r.x = 0.0f;            r.y = 0.0f; }
  out[idx] = r;
}

// One wave per 16-row M tile; NT 16-col N tiles held in v8f accumulators.
// D = (agg/cnt) @ W_l + root @ W_r (+bias, optional relu). K = 128 fixed.
template <int NT, int NOUT, bool RELU>
__global__ __launch_bounds__(256) void sage_wmma(const float* __restrict__ agg,
                                                 const float* __restrict__ cnt,
                                                 const float* __restrict__ root,
                                                 const v2f* __restrict__ Wlp,
                                                 const v2f* __restrict__ Wrp,
                                                 const float* __restrict__ bias,
                                                 float* __restrict__ out) {
  const int K = 128;
  int wid = (int)((blockIdx.x * blockDim.x + threadIdx.x) >> 5);
  int lane = threadIdx.x & 31;
  if (wid >= N_NODES / 16) return;             // wave-uniform (EXEC stays all-1s for WMMA)
  int mbase = wid * 16;
  int m = mbase + (lane & 15);
  int koff = (lane >> 4) << 1;                 // 0 for lanes 0-15, 2 for lanes 16-31
  float inv = 1.0f / fmaxf(cnt[m], 1.0f);      // mean normalization folded into A-fragment

  v8f acc[NT] = {};
  const float* aggRow  = agg  + (size_t)m * K;
  const float* rootRow = root + (size_t)m * K;

#pragma unroll 2
  for (int ks = 0; ks < K / 4; ++ks) {
    int kk = ks * 4 + koff;
    v2f am = *(const v2f*)(aggRow + kk);       // 8B aligned (kk even)
    am.x *= inv; am.y *= inv;
    v2f ax = *(const v2f*)(rootRow + kk);
#pragma unroll
    for (int nt = 0; nt < NT; ++nt) {
      v2f bl = Wlp[(ks * NT + nt) * 32 + lane];
      v2f br = Wrp[(ks * NT + nt) * 32 + lane];
      acc[nt] = __builtin_amdgcn_wmma_f32_16x16x4_f32(false, am, false, bl,
                                                      (short)0, acc[nt], false, false);
      acc[nt] = __builtin_amdgcn_wmma_f32_16x16x4_f32(false, ax, false, br,
                                                      (short)0, acc[nt], false, false);
    }
  }

  int colbase = lane & 15;
  int rowoff = (lane >> 4) * 8;
#pragma unroll
  for (int nt = 0; nt < NT; ++nt) {
    int col = nt * 16 + colbase;
    if (col < NOUT) {                          // masks padded columns (layer 2)
      float bv = bias[col];
#pragma unroll
      for (int r = 0; r < 8; ++r) {
        int row = mbase + rowoff + r;
        float v = acc[nt][r] + bv;
        if (RELU) v = fmaxf(v, 0.0f);
        out[(size_t)row * NOUT + col] = v;
      }
    }
  }
}

extern "C" void kernel_launch(void* const* d_in, const int* in_sizes, int n_in,
                              void* d_out, int out_size, void* d_ws, size_t ws_size,
                              hipStream_t stream) {
  (void)in_sizes; (void)n_in; (void)out_size; (void)ws_size;
  const float* x   = (const float*)d_in[0];
  const int*   ei  = (const int*)d_in[1];     // int32 (JAX x64 disabled)
  const float* W1l = (const float*)d_in[2];
  const float* W1r = (const float*)d_in[3];
  const float* b1  = (const float*)d_in[4];
  const float* W2l = (const float*)d_in[5];
  const float* W2r = (const float*)d_in[6];
  const float* b2  = (const float*)d_in[7];
  float* out = (float*)d_out;

  // Workspace layout (floats): agg[N*128] | h[N*128] | cnt[N] | packed weights
  float* ws  = (float*)d_ws;
  float* agg = ws;
  float* h   = agg + (size_t)N_NODES * HIDDEN;
  float* cnt = h   + (size_t)N_NODES * HIDDEN;
  v2f* W1lp = (v2f*)(cnt + N_NODES);          // 32 ksteps * 8 nt * 32 lanes = 8192 v2f
  v2f* W1rp = W1lp + 8192;
  v2f* W2lp = W1rp + 8192;                    // 32 ksteps * 3 nt * 32 lanes = 3072 v2f
  v2f* W2rp = W2lp + 3072;

  const int gemmBlocks = (N_NODES / 16 + 7) / 8;  // 8 waves/block, 6250 M-tiles -> 782

  zero_f32<<<2048, 256, 0, stream>>>(agg, (long long)N_NODES * HIDDEN);
  zero_f32<<<64,   256, 0, stream>>>(cnt, (long long)N_NODES);

  pack_w<<<32, 256, 0, stream>>>(W1l, IN_DIM, HIDDEN,  8, W1lp);
  pack_w<<<32, 256, 0, stream>>>(W1r, IN_DIM, HIDDEN,  8, W1rp);
  pack_w<<<12, 256, 0, stream>>>(W2l, HIDDEN, CLASSES, 3, W2lp);
  pack_w<<<12, 256, 0, stream>>>(W2r, HIDDEN, CLASSES, 3, W2rp);

  // Layer 1
  edge_scatter<<<N_EDGES / 8, 256, 0, stream>>>(x, ei, agg, cnt, 1);
  sage_wmma<8, HIDDEN, true><<<gemmBlocks, 256, 0, stream>>>(agg, cnt, x, W1lp, W1rp, b1, h);

  // Layer 2 (reuse agg; counts unchanged)
  zero_f32<<<2048, 256, 0, stream>>>(agg, (long long)N_NODES * HIDDEN);
  edge_scatter<<<N_EDGES / 8, 256, 0, stream>>>(h, ei, agg, cnt, 0);
  sage_wmma<3, CLASSES, false><<<gemmBlocks, 256, 0, stream>>>(agg, cnt, h, W2lp, W2rp, b2, out);
}